// AlignmentLoss_2327872274776
// MI455X (gfx1250) — compile-verified
//
#include <hip/hip_runtime.h>
#include <math.h>

// Problem constants (match reference)
#define PN   128   // N batches
#define PK   1024  // K inputs per batch
#define PM   512   // M phrases
#define PD   128   // D embedding dim
#define PP   4096  // P queries
#define MAXG 384   // upper bound on sum_b ceil(cnt_b/16) = (4096 + 128*15)/16 = 376
#define LSTR 132   // LDS row stride in floats (16B-aligned rows, conflict-free)

typedef __attribute__((ext_vector_type(2))) float v2f;
typedef __attribute__((ext_vector_type(8))) float v8f;

// ---------------------------------------------------------------- utilities
__global__ void k_zero(unsigned* p, int n) {
    int i = blockIdx.x * blockDim.x + threadIdx.x;
    int stride = gridDim.x * blockDim.x;
    for (; i < n; i += stride) p[i] = 0u;
}

// normalize phrase embeddings: Pn[m] = phrase[m] / max(||phrase[m]||, 1e-12)
__global__ void k_norm_phrase(const float* __restrict__ ph, float* __restrict__ Pn) {
    int m = blockIdx.x, d = threadIdx.x;  // 128 threads
    __shared__ float red[PD];
    float x = ph[m * PD + d];
    red[d] = x * x;
    __syncthreads();
    for (int s = PD / 2; s > 0; s >>= 1) {
        if (d < s) red[d] += red[d + s];
        __syncthreads();
    }
    float inv = 1.0f / fmaxf(sqrtf(red[0]), 1e-12f);
    Pn[m * PD + d] = x * inv;
}

// inverse norms of input embeddings (one block per (b,k) row)
__global__ void k_inv_input(const float* __restrict__ in, float* __restrict__ inv_in) {
    int row = blockIdx.x, d = threadIdx.x;  // 128 threads, N*K blocks
    __shared__ float red[PD];
    float x = in[(size_t)row * PD + d];
    red[d] = x * x;
    __syncthreads();
    for (int s = PD / 2; s > 0; s >>= 1) {
        if (d < s) red[d] += red[d + s];
        __syncthreads();
    }
    if (d == 0) inv_in[row] = 1.0f / fmaxf(sqrtf(red[0]), 1e-12f);
}

// histogram of p per batch, and the exact (b,m,k0) mask bitset
__global__ void k_hist_mask(const int* __restrict__ bidx, const int* __restrict__ midx,
                            const int* __restrict__ kidx, int* __restrict__ bin_cnt,
                            unsigned* __restrict__ maskw) {
    int p = blockIdx.x * blockDim.x + threadIdx.x;
    if (p >= PP) return;
    int b = bidx[p], m = midx[p], k = kidx[p];
    atomicAdd(&bin_cnt[b], 1);
    int bm = b * PM + m;
    atomicOr(&maskw[bm * (PK / 32) + (k >> 5)], 1u << (k & 31));
}

// serial prefix sum over 128 bins + build group table (groups of up to 16 p's per batch)
__global__ void k_prefix_groups(const int* __restrict__ bin_cnt, int* __restrict__ bin_start,
                                int* __restrict__ groups, int* __restrict__ numg) {
    if (blockIdx.x != 0 || threadIdx.x != 0) return;
    int acc = 0, ng = 0;
    for (int b = 0; b < PN; b++) {
        bin_start[b] = acc;
        int c = bin_cnt[b];
        for (int g = 0; g * 16 < c; g++) {
            int rem = c - g * 16;
            groups[ng * 3 + 0] = b;
            groups[ng * 3 + 1] = acc + g * 16;
            groups[ng * 3 + 2] = rem < 16 ? rem : 16;
            ng++;
        }
        acc += c;
    }
    bin_start[PN] = acc;
    *numg = ng;
}

// deterministic stable scatter: rank of p within bin = #{q < p : b_q == b_p}
__global__ void k_scatter(const int* __restrict__ bidx, const int* __restrict__ bin_start,
                          int* __restrict__ p_sorted) {
    int p = blockIdx.x * blockDim.x + threadIdx.x;
    if (p >= PP) return;
    int b = bidx[p], r = 0;
    for (int q = 0; q < p; q++) r += (bidx[q] == b) ? 1 : 0;
    p_sorted[bin_start[b] + r] = p;
}

// ---------------------------------------------------------------- main WMMA kernel
// One wave (32 threads) handles one group of <=16 p's sharing batch b.
__global__ __launch_bounds__(32) void k_main(
    const float* __restrict__ input, const float* __restrict__ Pn,
    const float* __restrict__ inv_in, const unsigned* __restrict__ maskw,
    const int* __restrict__ groups, const int* __restrict__ numg,
    const int* __restrict__ p_sorted, const int* __restrict__ midx,
    const int* __restrict__ kidx, const int* __restrict__ rneg,
    const float* __restrict__ temp_p, float* __restrict__ tri_arr,
    float* __restrict__ ce_arr) {
    int gi = blockIdx.x;
    if (gi >= *numg) return;
    int b = groups[gi * 3 + 0];
    int slot = groups[gi * 3 + 1];
    int cnt = groups[gi * 3 + 2];

    int l = threadIdx.x;
    int l15 = l & 15;
    int sh = (l < 16) ? 0 : 2;  // K-offset within 4-wide fragment per ISA A/B layout

    __shared__ float anchors[16 * LSTR];  // Pn rows for the 16 p's
    __shared__ float tile[16 * LSTR];     // streamed 16x128 operand tile
    __shared__ float posL[16 * LSTR];     // normalized positive embeddings
    __shared__ float outT[16 * 16];       // staged WMMA 16x16 result
    __shared__ int mS[16], kS[16];

    int p = -1, m = -1, k0 = -1, r0 = -1, r1 = -1;
    if (l < 16) {
        if (l < cnt) {
            p = p_sorted[slot + l];
            m = midx[p];
            k0 = kidx[p];
            r0 = rneg[p * 2 + 0];
            r1 = rneg[p * 2 + 1];
        }
        mS[l] = m;
        kS[l] = k0;
    }
    __syncthreads();

    // fill anchors (normalized phrase rows) and posL (normalized positives)
    for (int i = l; i < 16 * 32; i += 32) {  // 512 float4 slots
        int r = i >> 5, d4 = i & 31;
        int mm = mS[r], kk = kS[r];
        float4 av = make_float4(0.f, 0.f, 0.f, 0.f);
        float4 pv = make_float4(0.f, 0.f, 0.f, 0.f);
        if (mm >= 0) av = ((const float4*)(Pn + (size_t)mm * PD))[d4];
        if (kk >= 0) {
            float iv = inv_in[b * PK + kk];
            float4 t = ((const float4*)(input + ((size_t)b * PK + kk) * PD))[d4];
            pv = make_float4(t.x * iv, t.y * iv, t.z * iv, t.w * iv);
        }
        ((float4*)(anchors + r * LSTR))[d4] = av;
        ((float4*)(posL + r * LSTR))[d4] = pv;
    }
    __syncthreads();

    // Preload A fragments (anchors) into registers: step s covers d = 4s..4s+3
    v2f frag[32];
#pragma unroll
    for (int s = 0; s < 32; s++)
        frag[s] = *(const v2f*)(anchors + l15 * LSTR + s * 4 + sh);

    float t0 = -INFINITY, t1 = -INFINITY, t2 = -INFINITY;
    float sPos = 0.f, sR0 = 0.f, sR1 = 0.f;
    int bm = b * PM + ((m >= 0) ? m : 0);

    // ---------------- Part 1: sim row over K, online top-3 hard negatives ----------------
    for (int kt = 0; kt < PK / 16; kt++) {
        int kbase = kt * 16;
        for (int i = l; i < 16 * 32; i += 32) {
            int r = i >> 5, d4 = i & 31;
            int kk = kbase + r;
            float iv = inv_in[b * PK + kk];
            float4 t = ((const float4*)(input + ((size_t)b * PK + kk) * PD))[d4];
            ((float4*)(tile + r * LSTR))[d4] = make_float4(t.x * iv, t.y * iv, t.z * iv, t.w * iv);
        }
        __syncthreads();
        v8f acc = {};
#pragma unroll
        for (int s = 0; s < 32; s++) {
            v2f bfr = *(const v2f*)(tile + l15 * LSTR + s * 4 + sh);
            acc = __builtin_amdgcn_wmma_f32_16x16x4_f32(false, frag[s], false, bfr,
                                                        (short)0, acc, false, false);
        }
        int rowbase = (l < 16) ? 0 : 8;
#pragma unroll
        for (int r = 0; r < 8; r++) outT[(rowbase + r) * 16 + l15] = acc[r];
        __syncthreads();
        if (l < 16) {
            unsigned w = maskw[bm * (PK / 32) + (kbase >> 5)];
            int bit0 = kbase & 31;
#pragma unroll
            for (int c = 0; c < 16; c++) {
                float v = outT[l * 16 + c];
                int k = kbase + c;
                if (k == k0) sPos = v;
                if (k == r0) sR0 = v;
                if (k == r1) sR1 = v;
                if (!((w >> (bit0 + c)) & 1u)) {
                    if (v > t0)      { t2 = t1; t1 = t0; t0 = v; }
                    else if (v > t1) { t2 = t1; t1 = v; }
                    else if (v > t2) { t2 = v; }
                }
            }
        }
        __syncthreads();
    }

    if (l < 16 && l < cnt) {
        float tri = fmaxf(t0 - sPos + 1.f, 0.f) + fmaxf(t1 - sPos + 1.f, 0.f) +
                    fmaxf(t2 - sPos + 1.f, 0.f) + fmaxf(sR0 - sPos + 1.f, 0.f) +
                    fmaxf(sR1 - sPos + 1.f, 0.f);
        tri_arr[p] = tri;
    }

    // ---------------- Part 2: CE over M phrases with online logsumexp ----------------
    float temp = *temp_p;
#pragma unroll
    for (int s = 0; s < 32; s++)  // B fragments = positives (constant across m-tiles)
        frag[s] = *(const v2f*)(posL + l15 * LSTR + s * 4 + sh);

    float runmax = -INFINITY, runsum = 0.f, target = 0.f;
    for (int mt = 0; mt < PM / 16; mt++) {
        int mbase = mt * 16;
        for (int i = l; i < 16 * 32; i += 32) {
            int r = i >> 5, d4 = i & 31;
            ((float4*)(tile + r * LSTR))[d4] =
                ((const float4*)(Pn + (size_t)(mbase + r) * PD))[d4];
        }
        __syncthreads();
        v8f acc = {};
#pragma unroll
        for (int s = 0; s < 32; s++) {
            v2f afr = *(const v2f*)(tile + l15 * LSTR + s * 4 + sh);
            acc = __builtin_amdgcn_wmma_f32_16x16x4_f32(false, afr, false, frag[s],
                                                        (short)0, acc, false, false);
        }
        int rowbase = (l < 16) ? 0 : 8;
#pragma unroll
        for (int r = 0; r < 8; r++) outT[(rowbase + r) * 16 + l15] = acc[r];
        __syncthreads();
        if (l < 16) {
#pragma unroll
            for (int r = 0; r < 16; r++) {
                float t = outT[r * 16 + l] * temp;  // column l = this p
                float nm = fmaxf(runmax, t);
                runsum = runsum * expf(runmax - nm) + expf(t - nm);
                runmax = nm;
                if (mbase + r == m) target = t;
            }
        }
        __syncthreads();
    }
    if (l < 16 && l < cnt) ce_arr[p] = (logf(runsum) + runmax) - target;
}

// fixed-order deterministic reduction of the per-p losses
__global__ void k_finalize(const float* __restrict__ tri_arr, const float* __restrict__ ce_arr,
                           float* __restrict__ out) {
    __shared__ float s1[256], s2[256];
    int t = threadIdx.x;
    float a = 0.f, c = 0.f;
    for (int i = t; i < PP; i += 256) { a += tri_arr[i]; c += ce_arr[i]; }
    s1[t] = a; s2[t] = c;
    __syncthreads();
    for (int s = 128; s > 0; s >>= 1) {
        if (t < s) { s1[t] += s1[t + s]; s2[t] += s2[t + s]; }
        __syncthreads();
    }
    if (t == 0) {
        out[0] = s1[0] / (PP * 5.0f);
        out[1] = s2[0] / (float)PP;
    }
}

extern "C" void kernel_launch(void* const* d_in, const int* in_sizes, int n_in,
                              void* d_out, int out_size, void* d_ws, size_t ws_size,
                              hipStream_t stream) {
    const float* phrase = (const float*)d_in[0];  // (M, D)
    const float* input  = (const float*)d_in[1];  // (N, K, D)
    const int* bidx = (const int*)d_in[2];        // (P,)
    const int* midx = (const int*)d_in[3];        // (P,)
    const int* kidx = (const int*)d_in[4];        // (P,)
    const int* rneg = (const int*)d_in[5];        // (P, 2)
    const float* temp = (const float*)d_in[6];    // scalar
    float* out = (float*)d_out;                   // (triplet, ce)

    // workspace carve (~9.3 MB)
    float* Pn = (float*)d_ws;                                  // M*D
    float* inv_in = Pn + PM * PD;                              // N*K
    unsigned* maskw = (unsigned*)(inv_in + PN * PK);           // N*M*K/32
    int* bin_cnt = (int*)(maskw + (size_t)PN * PM * (PK / 32));// N  (contiguous after maskw)
    int* bin_start = bin_cnt + PN;                             // N+1 (pad to N+4)
    int* groups = bin_start + (PN + 4);                        // 3*MAXG
    int* numg = groups + 3 * MAXG;                             // 1 (pad 4)
    int* p_sorted = numg + 4;                                  // P
    float* tri_arr = (float*)(p_sorted + PP);                  // P
    float* ce_arr = tri_arr + PP;                              // P

    k_zero<<<1024, 256, 0, stream>>>(maskw, PN * PM * (PK / 32) + PN);
    k_norm_phrase<<<PM, PD, 0, stream>>>(phrase, Pn);
    k_inv_input<<<PN * PK, PD, 0, stream>>>(input, inv_in);
    k_hist_mask<<<(PP + 255) / 256, 256, 0, stream>>>(bidx, midx, kidx, bin_cnt, maskw);
    k_prefix_groups<<<1, 1, 0, stream>>>(bin_cnt, bin_start, groups, numg);
    k_scatter<<<(PP + 255) / 256, 256, 0, stream>>>(bidx, bin_start, p_sorted);
    k_main<<<MAXG, 32, 0, stream>>>(input, Pn, inv_in, maskw, groups, numg, p_sorted,
                                    midx, kidx, rneg, temp, tri_arr, ce_arr);
    k_finalize<<<1, 256, 0, stream>>>(tri_arr, ce_arr, out);
}